// LSTMA_42855183679638
// MI455X (gfx1250) — compile-verified
//
#include <hip/hip_runtime.h>
#include <math.h>

typedef __attribute__((ext_vector_type(2))) float v2f;
typedef __attribute__((ext_vector_type(8))) float v8f;

#define T_SZ   131072
#define E_SZ   512
#define H_SZ   512
#define K_SZ   1024
#define L_SZ   128
#define XN     1025      // 2E+1
#define PREDN  1537      // E+2H+1

// ---- workspace layout (bytes) ----
#define WS_ALPHA 0u        // T floats            (524288)
#define WS_VALS  524288u   // K floats            (4096)
#define WS_IDX   528384u   // K ints              (4096)
#define WS_A     532480u   // K floats            (4096)
#define WS_ATTN  536576u   // H floats            (2048)
#define WS_V     538624u   // E floats            (2048)
#define WS_GI    540672u   // 1536 floats         (6144)
#define WS_GH    546816u   // 1536 floats         (6144)
#define WS_HIST  552960u   // 256 u32             (1024)
#define WS_SEL   553984u   // sel[0]=prefix sel[1]=needed (64)
#define WS_TCG   554048u   // 4096 u32 counts >th (16384)
#define WS_TCE   570432u   // 4096 u32 counts ==th(16384)
#define WS_TOG   586816u   // 4096 u32 offsets    (16384)
#define WS_TOE   603200u   // 4096 u32 offsets    (16384)
#define WS_PART  619584u   // 32*512 floats       (65536)

__device__ __forceinline__ unsigned f2key(float f) {
  unsigned u = __float_as_uint(f);
  return (u & 0x80000000u) ? ~u : (u | 0x80000000u);
}

// ---------------- init: zero radix state ----------------
__global__ void k_init(unsigned* hist, unsigned* sel) {
  int t = threadIdx.x;
  hist[t] = 0u;                       // 256 threads
  if (t == 0) { sel[0] = 0u; sel[1] = (unsigned)K_SZ; }
}

// ---------------- v = mean(emb[topic]) ; append to out_vs ----------------
__global__ void k_topic_mean(const int* __restrict__ topic, const float* __restrict__ emb,
                             float* __restrict__ v, float* __restrict__ out_vs_last) {
  int e = threadIdx.x;                // 512 threads
  float acc = 0.f;
  for (int l = 0; l < L_SZ; ++l)
    acc += emb[(size_t)topic[l] * E_SZ + e];
  float val = acc * (1.0f / (float)L_SZ);
  v[e] = val;
  out_vs_last[e] = val;
}

// ---------------- fused: alpha = vs @ v  (WMMA f32 16x16x4)  +  copy vs -> out ----------------
// One wave per 16 rows. A: 16x4 f32 chunks of vs rows; B: v chunk replicated across N.
// D[m][n] == dot(vs_row_m, v) for every n; extract N=0 from lanes 0 and 16.
__global__ void k_alpha_copy(const float* __restrict__ vs, const float* __restrict__ v,
                             float* __restrict__ out_vs, float* __restrict__ alpha) {
  int lane = threadIdx.x & 31;
  int wave = threadIdx.x >> 5;
  int row0 = (blockIdx.x * 8 + wave) * 16;
  int m  = lane & 15;
  int kk = (lane >> 4) << 1;          // 0 or 2
  const float* arow = vs     + (size_t)(row0 + m) * E_SZ + kk;
  float*       orow = out_vs + (size_t)(row0 + m) * E_SZ + kk;  // 4B-aligned only (out base +513)
  v8f acc = {};
  for (int kb = 0; kb < E_SZ; kb += 4) {
    v2f a2 = *(const v2f*)(arow + kb);        // A[m][kk..kk+1] of chunk kb  (8B aligned)
    v2f b2 = *(const v2f*)(v + kb + kk);      // B[kk..kk+1][*] = v (L2-hot)
    orow[kb]     = a2.x;                      // copy (scalar stores: dest 4B-aligned)
    orow[kb + 1] = a2.y;
    acc = __builtin_amdgcn_wmma_f32_16x16x4_f32(
        /*neg_a=*/false, a2, /*neg_b=*/false, b2,
        /*c_mod=*/(short)0, acc, /*reuse_a=*/false, /*reuse_b=*/false);
  }
  if (lane == 0) {
    float* ap = alpha + row0;
    ap[0]=acc[0]; ap[1]=acc[1]; ap[2]=acc[2]; ap[3]=acc[3];
    ap[4]=acc[4]; ap[5]=acc[5]; ap[6]=acc[6]; ap[7]=acc[7];
  } else if (lane == 16) {
    float* ap = alpha + row0 + 8;
    ap[0]=acc[0]; ap[1]=acc[1]; ap[2]=acc[2]; ap[3]=acc[3];
    ap[4]=acc[4]; ap[5]=acc[5]; ap[6]=acc[6]; ap[7]=acc[7];
  }
}

// ---------------- radix select (MSB-first, 4 passes of 8 bits) ----------------
__global__ void k_hist(const float* __restrict__ alpha, const unsigned* __restrict__ sel,
                       unsigned* __restrict__ hist, int pass) {
  unsigned prefix = sel[0];
  unsigned maskHi = (pass == 0) ? 0u : (0xFFFFFFFFu << (32 - 8 * pass));
  int shift = 24 - 8 * pass;
  int i = blockIdx.x * blockDim.x + threadIdx.x;
  int stride = gridDim.x * blockDim.x;
  for (; i < T_SZ; i += stride) {
    unsigned key = f2key(alpha[i]);
    if ((key & maskHi) == prefix)
      atomicAdd(&hist[(key >> shift) & 0xFFu], 1u);
  }
}

__global__ void k_scan(unsigned* __restrict__ sel, unsigned* __restrict__ hist, int pass) {
  if (threadIdx.x != 0) return;
  int shift = 24 - 8 * pass;
  unsigned needed = sel[1];
  unsigned cum = 0;
  for (int d = 255; d >= 0; --d) {
    unsigned c = hist[d];
    if (cum + c >= needed) {
      sel[0] |= ((unsigned)d) << shift;
      sel[1] = needed - cum;            // rank within the ==digit class
      break;
    }
    cum += c;
  }
  for (int d = 0; d < 256; ++d) hist[d] = 0u;  // reset for next pass
}

// ---------------- deterministic compaction (counts -> block scan -> ordered write) ----------------
#define CCHUNK 32   // 4096 threads * 32 elems = T
__global__ void k_count(const float* __restrict__ alpha, const unsigned* __restrict__ sel,
                        unsigned* __restrict__ tcg, unsigned* __restrict__ tce) {
  int g = blockIdx.x * blockDim.x + threadIdx.x;   // 0..4095
  unsigned th = sel[0];
  unsigned cg = 0, ce = 0;
  int t0 = g * CCHUNK;
  for (int j = 0; j < CCHUNK; ++j) {
    unsigned key = f2key(alpha[t0 + j]);
    cg += (key > th) ? 1u : 0u;
    ce += (key == th) ? 1u : 0u;
  }
  tcg[g] = cg; tce[g] = ce;
}

__global__ void k_scanoff(const unsigned* __restrict__ tcg, const unsigned* __restrict__ tce,
                          unsigned* __restrict__ tog, unsigned* __restrict__ toe) {
  __shared__ unsigned sg[1024], se[1024];
  int t = threadIdx.x;                 // 1024 threads, 4 entries each
  unsigned lg[4], le[4], sumg = 0, sume = 0;
  for (int j = 0; j < 4; ++j) {
    lg[j] = tcg[t * 4 + j]; sumg += lg[j];
    le[j] = tce[t * 4 + j]; sume += le[j];
  }
  sg[t] = sumg; se[t] = sume;
  __syncthreads();
  for (int off = 1; off < 1024; off <<= 1) {
    unsigned ag = (t >= off) ? sg[t - off] : 0u;
    unsigned ae = (t >= off) ? se[t - off] : 0u;
    __syncthreads();
    sg[t] += ag; se[t] += ae;
    __syncthreads();
  }
  unsigned excg = (t == 0) ? 0u : sg[t - 1];
  unsigned exce = (t == 0) ? 0u : se[t - 1];
  for (int j = 0; j < 4; ++j) {
    tog[t * 4 + j] = excg; excg += lg[j];
    toe[t * 4 + j] = exce; exce += le[j];
  }
}

__global__ void k_write(const float* __restrict__ alpha, const unsigned* __restrict__ sel,
                        const unsigned* __restrict__ tog, const unsigned* __restrict__ toe,
                        float* __restrict__ vals, int* __restrict__ idx) {
  int g = blockIdx.x * blockDim.x + threadIdx.x;   // 0..4095
  unsigned th = sel[0];
  unsigned needed_eq = sel[1];
  unsigned base_eq = (unsigned)K_SZ - needed_eq;   // # strictly-greater elements
  unsigned og = tog[g], oe = toe[g];
  int t0 = g * CCHUNK;
  for (int j = 0; j < CCHUNK; ++j) {
    int t = t0 + j;
    float av = alpha[t];
    unsigned key = f2key(av);
    if (key > th) {
      vals[og] = av; idx[og] = t; ++og;
    } else if (key == th) {
      if (oe < needed_eq) { vals[base_eq + oe] = av; idx[base_eq + oe] = t; }
      ++oe;                                        // lowest-index ties win (matches jax)
    }
  }
}

// ---------------- softmax over K=1024 ----------------
__global__ void k_softmax(const float* __restrict__ vals, float* __restrict__ a) {
  __shared__ float sm[1024];
  int t = threadIdx.x;
  float v = vals[t];
  sm[t] = v; __syncthreads();
  for (int s = 512; s > 0; s >>= 1) { if (t < s) sm[t] = fmaxf(sm[t], sm[t + s]); __syncthreads(); }
  float mx = sm[0]; __syncthreads();
  float e = expf(v - mx);
  sm[t] = e; __syncthreads();
  for (int s = 512; s > 0; s >>= 1) { if (t < s) sm[t] += sm[t + s]; __syncthreads(); }
  a[t] = e / sm[0];
}

// ---------------- attn_h = a @ hs[idx]  (32 partial blocks, no atomics) ----------------
__global__ void k_attn(const float* __restrict__ a, const int* __restrict__ idx,
                       const float* __restrict__ hs, float* __restrict__ part) {
  int t = threadIdx.x;                 // 0..511
  int kb = blockIdx.x * 32;            // 32 blocks
  float acc = 0.f;
  for (int j = 0; j < 32; ++j) {
    int k = kb + j;
    float w = a[k];
    acc = fmaf(w, hs[(size_t)idx[k] * H_SZ + t], acc);
  }
  part[blockIdx.x * H_SZ + t] = acc;
}

__global__ void k_attn_red(const float* __restrict__ part, float* __restrict__ attn) {
  int t = threadIdx.x;                 // 512
  float acc = 0.f;
  for (int b = 0; b < 32; ++b) acc += part[b * H_SZ + t];
  attn[t] = acc;
}

// ---------------- score = [v, attn, h, K] . w_score + b ----------------
__global__ void k_score(const float* __restrict__ v, const float* __restrict__ attn,
                        const float* __restrict__ h, const float* __restrict__ w_score,
                        const float* __restrict__ b_score, float* __restrict__ out) {
  __shared__ float sm[512];
  int t = threadIdx.x;
  float acc = 0.f;
  for (int i = t; i < PREDN; i += 512) {
    float pv = (i < E_SZ) ? v[i]
             : (i < E_SZ + H_SZ) ? attn[i - E_SZ]
             : (i < E_SZ + 2 * H_SZ) ? h[i - E_SZ - H_SZ]
             : (float)K_SZ;
    acc = fmaf(pv, w_score[i], acc);
  }
  sm[t] = acc; __syncthreads();
  for (int s = 256; s > 0; s >>= 1) { if (t < s) sm[t] += sm[t + s]; __syncthreads(); }
  if (t == 0) out[0] = sm[0] + b_score[0];
}

// ---------------- GRU gate mat-vecs: one wave per output row ----------------
__global__ void k_gates(const float* __restrict__ v, const float* __restrict__ score_s,
                        const float* __restrict__ h, const float* __restrict__ w_ih,
                        const float* __restrict__ w_hh, const float* __restrict__ b_ih,
                        const float* __restrict__ b_hh, float* __restrict__ gi,
                        float* __restrict__ gh) {
  int lane = threadIdx.x & 31;
  int wid = blockIdx.x * (blockDim.x >> 5) + (threadIdx.x >> 5);  // 0..3071
  float ss = score_s[0];
  float pos = (ss >= 0.5f) ? 1.0f : 0.0f;
  float acc = 0.f;
  if (wid < 3 * H_SZ) {
    const float* wr = w_ih + (size_t)wid * XN;
    for (int c = lane; c < XN; c += 32) {
      float x = (c < E_SZ) ? v[c] * pos
              : (c < 2 * E_SZ) ? v[c - E_SZ] * (1.0f - pos)
              : ss;
      acc = fmaf(wr[c], x, acc);
    }
    for (int off = 16; off > 0; off >>= 1) acc += __shfl_down(acc, off, 32);
    if (lane == 0) gi[wid] = acc + b_ih[wid];
  } else {
    int r = wid - 3 * H_SZ;
    const float* wr = w_hh + (size_t)r * H_SZ;
    for (int c = lane; c < H_SZ; c += 32) acc = fmaf(wr[c], h[c], acc);
    for (int off = 16; off > 0; off >>= 1) acc += __shfl_down(acc, off, 32);
    if (lane == 0) gh[r] = acc + b_hh[r];
  }
}

// ---------------- GRU pointwise update; write h_new + append to out_hs ----------------
__global__ void k_update(const float* __restrict__ gi, const float* __restrict__ gh,
                         const float* __restrict__ h, float* __restrict__ out_h,
                         float* __restrict__ out_hs_last) {
  int t = threadIdx.x;                 // 512
  float r = 1.0f / (1.0f + expf(-(gi[t] + gh[t])));
  float z = 1.0f / (1.0f + expf(-(gi[H_SZ + t] + gh[H_SZ + t])));
  float n = tanhf(gi[2 * H_SZ + t] + r * gh[2 * H_SZ + t]);
  float hn = (1.0f - z) * n + z * h[t];
  out_h[t] = hn;
  out_hs_last[t] = hn;
}

// ---------------- bulk copy hs -> out_hs (scalar b32: dest offset is odd-float) ----------------
__global__ void k_copy_hs(const float* __restrict__ src, float* __restrict__ dst, int n) {
  int i = blockIdx.x * blockDim.x + threadIdx.x;
  int stride = gridDim.x * blockDim.x;
  for (; i < n; i += stride) dst[i] = src[i];
}

extern "C" void kernel_launch(void* const* d_in, const int* in_sizes, int n_in,
                              void* d_out, int out_size, void* d_ws, size_t ws_size,
                              hipStream_t stream) {
  (void)in_sizes; (void)n_in; (void)out_size; (void)ws_size;
  const int*   topic   = (const int*)  d_in[0];
  const float* score_s = (const float*)d_in[1];
  const float* emb     = (const float*)d_in[2];
  const float* h       = (const float*)d_in[3];
  const float* vs      = (const float*)d_in[4];
  const float* hs      = (const float*)d_in[5];
  const float* w_ih    = (const float*)d_in[6];
  const float* w_hh    = (const float*)d_in[7];
  const float* b_ih    = (const float*)d_in[8];
  const float* b_hh    = (const float*)d_in[9];
  const float* w_score = (const float*)d_in[10];
  const float* b_score = (const float*)d_in[11];
  float* out = (float*)d_out;

  char* ws = (char*)d_ws;
  float*    alpha = (float*)   (ws + WS_ALPHA);
  float*    vals  = (float*)   (ws + WS_VALS);
  int*      idx   = (int*)     (ws + WS_IDX);
  float*    aarr  = (float*)   (ws + WS_A);
  float*    attn  = (float*)   (ws + WS_ATTN);
  float*    vvec  = (float*)   (ws + WS_V);
  float*    gi    = (float*)   (ws + WS_GI);
  float*    gh    = (float*)   (ws + WS_GH);
  unsigned* hist  = (unsigned*)(ws + WS_HIST);
  unsigned* sel   = (unsigned*)(ws + WS_SEL);
  unsigned* tcg   = (unsigned*)(ws + WS_TCG);
  unsigned* tce   = (unsigned*)(ws + WS_TCE);
  unsigned* tog   = (unsigned*)(ws + WS_TOG);
  unsigned* toe   = (unsigned*)(ws + WS_TOE);
  float*    part  = (float*)   (ws + WS_PART);

  // output layout: score[1], h_new[512], vs_new[(T+1)*E], hs_new[(T+1)*H]
  float* out_vs = out + 1 + H_SZ;                                    // +513
  float* out_hs = out_vs + (size_t)(T_SZ + 1) * E_SZ;

  k_init      <<<1, 256, 0, stream>>>(hist, sel);
  k_topic_mean<<<1, 512, 0, stream>>>(topic, emb, vvec, out_vs + (size_t)T_SZ * E_SZ);
  k_alpha_copy<<<1024, 256, 0, stream>>>(vs, vvec, out_vs, alpha);   // WMMA + vs copy
  for (int p = 0; p < 4; ++p) {
    k_hist<<<256, 512, 0, stream>>>(alpha, sel, hist, p);
    k_scan<<<1, 1, 0, stream>>>(sel, hist, p);
  }
  k_count  <<<8, 512, 0, stream>>>(alpha, sel, tcg, tce);
  k_scanoff<<<1, 1024, 0, stream>>>(tcg, tce, tog, toe);
  k_write  <<<8, 512, 0, stream>>>(alpha, sel, tog, toe, vals, idx);
  k_softmax<<<1, 1024, 0, stream>>>(vals, aarr);
  k_attn   <<<32, 512, 0, stream>>>(aarr, idx, hs, part);
  k_attn_red<<<1, 512, 0, stream>>>(part, attn);
  k_score  <<<1, 512, 0, stream>>>(vvec, attn, h, w_score, b_score, out);
  k_gates  <<<384, 256, 0, stream>>>(vvec, score_s, h, w_ih, w_hh, b_ih, b_hh, gi, gh);
  k_update <<<1, 512, 0, stream>>>(gi, gh, h, out + 1, out_hs + (size_t)T_SZ * H_SZ);
  k_copy_hs<<<4096, 256, 0, stream>>>(hs, out_hs, T_SZ * H_SZ);
}